// UnfoldLayer_9363028706226
// MI455X (gfx1250) — compile-verified
//
#include <hip/hip_runtime.h>
#include <stdint.h>

// Unfold (im2col): x[32,64,64,64] f32 -> out[32, 64*9, 62*62] f32.
// Pure data movement: ~317 MB total traffic -> ~13.6 us floor @ 23.3 TB/s.
// Strategy: one workgroup per (b,c) 64x64 plane. Stage the plane into LDS
// once via CDNA5 async global->LDS copies (ASYNCcnt path), then emit the 9
// shifted copies as coalesced float4 non-temporal stores, serving the 9x
// element reuse from LDS instead of re-reading L2/HBM.

#define BATCH   32
#define CH      64
#define H       64
#define W       64
#define HO      62
#define WO      62
#define LLEN    (HO * WO)     // 3844
#define L4      (LLEN / 4)    // 961 (exact: 3844 = 4*961)
#define PLANE   (H * W)       // 4096 floats = 16 KB
#define THREADS 256

typedef float v4f __attribute__((ext_vector_type(4)));

__global__ __launch_bounds__(THREADS)
void unfold_kernel(const float* __restrict__ x, float* __restrict__ out) {
    __shared__ __align__(16) float smem[PLANE];

    const int plane = blockIdx.x;      // plane = b*CH + c
    const int tid   = threadIdx.x;

    // ---------- Stage: async copy one 64x64 plane (16 KB) into LDS ----------
    {
        const float* src  = x + (size_t)plane * PLANE;
        uint32_t     lds0 = (uint32_t)(uintptr_t)&smem[0];   // low 32 bits of a
                                                             // flat shared addr
                                                             // == LDS byte offset
        #pragma unroll
        for (int it = 0; it < PLANE / 4 / THREADS; ++it) {   // 4 iterations
            int      idx   = tid + it * THREADS;             // float4 index
            uint32_t loff  = lds0 + (uint32_t)idx * 16u;
            uint64_t gaddr = (uint64_t)(uintptr_t)src + (uint64_t)idx * 16u;
            asm volatile("global_load_async_to_lds_b128 %0, %1, off"
                         :: "v"(loff), "v"(gaddr)
                         : "memory");
        }
#if __has_builtin(__builtin_amdgcn_s_wait_asynccnt)
        __builtin_amdgcn_s_wait_asynccnt(0);
#else
        asm volatile("s_wait_asynccnt 0" ::: "memory");
#endif
    }
    __syncthreads();   // all waves' async copies complete -> LDS plane valid

    // ---------- Emit: 9 shifted copies, float4 NT stores ----------
    // out row base for this plane: (b*576 + c*9) * LLEN == 9*plane*LLEN
    float* obase = out + (size_t)plane * 9u * (size_t)LLEN;

    for (int q = tid; q < L4; q += THREADS) {
        const int l  = q * 4;          // position within the 62x62 grid
        const int i0 = l / WO;         // magic-number div (compile-time const)
        const int j0 = l - i0 * WO;

        // LDS base indices for the 4 consecutive output positions; computed
        // once and reused across all 9 kernel offsets.
        int lid[4];
        #pragma unroll
        for (int e = 0; e < 4; ++e) {
            int j = j0 + e, i = i0;
            if (j >= WO) { j -= WO; i += 1; }   // row wrap in the 62-wide grid
            lid[e] = i * W + j;
        }

        #pragma unroll
        for (int k = 0; k < 9; ++k) {
            const int ki  = k / 3;              // compile-time constants
            const int kj  = k % 3;
            const int off = ki * W + kj;
            v4f v;
            v.x = smem[lid[0] + off];
            v.y = smem[lid[1] + off];
            v.z = smem[lid[2] + off];
            v.w = smem[lid[3] + off];
            // Streaming output: non-temporal so 283 MB of writes bypass
            // temporal residency in L2.
            __builtin_nontemporal_store(v, (v4f*)(obase + (size_t)k * LLEN + l));
        }
    }
}

extern "C" void kernel_launch(void* const* d_in, const int* in_sizes, int n_in,
                              void* d_out, int out_size, void* d_ws, size_t ws_size,
                              hipStream_t stream) {
    const float* x   = (const float*)d_in[0];
    float*       out = (float*)d_out;
    unfold_kernel<<<BATCH * CH, THREADS, 0, stream>>>(x, out);
}